// WorkerLSTM_91104846283103
// MI455X (gfx1250) — compile-verified
//
#include <hip/hip_runtime.h>
#include <hip/hip_bf16.h>

#define BG    16        // graphs
#define NPER  1000
#define BN    16000     // nodes
#define NE    128000    // edges
#define ND    8
#define HD    256
#define ED    5
#define HEADS 4
#define HH    (HEADS*HD)   // 1024
#define GEPS  1e-5f
#define SLOPE 0.2f

typedef __attribute__((ext_vector_type(2))) float v2f;
typedef __attribute__((ext_vector_type(8))) float v8f;

// ---------- helpers ----------
__device__ __forceinline__ unsigned f2ord(float f) {
  unsigned u = __float_as_uint(f);
  return (u & 0x80000000u) ? ~u : (u | 0x80000000u);
}
__device__ __forceinline__ float ord2f(unsigned u) {
  return (u & 0x80000000u) ? __uint_as_float(u ^ 0x80000000u) : __uint_as_float(~u);
}
__device__ __forceinline__ float sigm(float x) { return 1.0f / (1.0f + expf(-x)); }
__device__ __forceinline__ v2f ld2(const float* __restrict__ p) {
  v2f r; r.x = p[0]; r.y = p[1]; return r;
}

// ---------- WMMA GEMM: C[M][N] = A[M][lda] @ W[N][ldw]^T (+bias) (+addend[idx[m]]) (+relu)
// f32 path via V_WMMA_F32_16X16X4_F32. One wave = one 16x64 output tile
// (4 accumulators; A fragment reused across 4 WMMAs per K-step; fragments
// double-buffered so next chunk's loads overlap current chunk's WMMAs).
__global__ __launch_bounds__(256) void gemm_wmma_f32(
    const float* __restrict__ A, int lda,
    const float* __restrict__ W, int ldw,
    const float* __restrict__ bias,
    const float* __restrict__ addend, const int* __restrict__ addIdx,
    float* __restrict__ C, int M, int N, int K, int applyRelu)
{
  const int wave = threadIdx.x >> 5;
  const int lane = threadIdx.x & 31;
  const int ntn  = N >> 6;                  // 64-wide N supertiles
  const int tile = blockIdx.x * 8 + wave;
  const int mt = tile / ntn;
  const int nb = tile % ntn;
  if (mt * 16 >= M) return;                 // wave-uniform exit (EXEC stays all-ones)

  const int r16  = lane & 15;
  const int hi   = lane >> 4;
  const int koff = hi * 2;                  // K-half split: lanes16-31 hold K+2,K+3
  const float* __restrict__ Arow  = A + (size_t)(mt * 16 + r16) * lda + koff;
  const float* __restrict__ Wb0   = W + (size_t)(nb * 64 + r16) * ldw + koff;
  const float* __restrict__ Wb1   = Wb0 + (size_t)16 * ldw;
  const float* __restrict__ Wb2   = Wb0 + (size_t)32 * ldw;
  const float* __restrict__ Wb3   = Wb0 + (size_t)48 * ldw;

  v8f acc0 = {}, acc1 = {}, acc2 = {}, acc3 = {};

  // preload chunk 0
  v2f a_c  = ld2(Arow);
  v2f b0_c = ld2(Wb0), b1_c = ld2(Wb1), b2_c = ld2(Wb2), b3_c = ld2(Wb3);

  for (int k0 = 0; k0 + 4 < K; k0 += 4) {
    // issue next chunk's loads first so they overlap the WMMAs below
    const int kn = k0 + 4;
    v2f a_n  = ld2(Arow + kn);
    v2f b0_n = ld2(Wb0 + kn), b1_n = ld2(Wb1 + kn),
        b2_n = ld2(Wb2 + kn), b3_n = ld2(Wb3 + kn);

    acc0 = __builtin_amdgcn_wmma_f32_16x16x4_f32(false, a_c, false, b0_c, (short)0, acc0, false, false);
    acc1 = __builtin_amdgcn_wmma_f32_16x16x4_f32(false, a_c, false, b1_c, (short)0, acc1, false, false);
    acc2 = __builtin_amdgcn_wmma_f32_16x16x4_f32(false, a_c, false, b2_c, (short)0, acc2, false, false);
    acc3 = __builtin_amdgcn_wmma_f32_16x16x4_f32(false, a_c, false, b3_c, (short)0, acc3, false, false);

    a_c = a_n; b0_c = b0_n; b1_c = b1_n; b2_c = b2_n; b3_c = b3_n;
  }
  // last chunk
  acc0 = __builtin_amdgcn_wmma_f32_16x16x4_f32(false, a_c, false, b0_c, (short)0, acc0, false, false);
  acc1 = __builtin_amdgcn_wmma_f32_16x16x4_f32(false, a_c, false, b1_c, (short)0, acc1, false, false);
  acc2 = __builtin_amdgcn_wmma_f32_16x16x4_f32(false, a_c, false, b2_c, (short)0, acc2, false, false);
  acc3 = __builtin_amdgcn_wmma_f32_16x16x4_f32(false, a_c, false, b3_c, (short)0, acc3, false, false);

  const int rbase = mt * 16 + hi * 8;       // C/D: VGPR i -> row i (lanes 0-15) / row i+8
  int ai[8];
  if (addend) {
    #pragma unroll
    for (int i = 0; i < 8; i++) ai[i] = addIdx[rbase + i];
  }
  v8f accs[4] = {acc0, acc1, acc2, acc3};
  #pragma unroll
  for (int j = 0; j < 4; j++) {
    const int col = nb * 64 + j * 16 + r16;
    const float bv = bias ? bias[col] : 0.0f;
    #pragma unroll
    for (int i = 0; i < 8; i++) {
      const int row = rbase + i;
      float v = accs[j][i] + bv;
      if (addend) v += addend[(size_t)ai[i] * N + col];
      if (applyRelu) v = fmaxf(v, 0.0f);
      C[(size_t)row * N + col] = v;
    }
  }
}

// ---------- edge pass 1: logits + segment max (one wave per (edge, head)) ----------
__global__ __launch_bounds__(256) void edge_logits_k(
    const float* __restrict__ xl, const float* __restrict__ xr,
    const float* __restrict__ eattr, const float* __restrict__ We,
    const float* __restrict__ att,
    const int* __restrict__ src, const int* __restrict__ dst,
    float* __restrict__ logit, unsigned* __restrict__ mx)
{
  const int wid = (int)((blockIdx.x * blockDim.x + threadIdx.x) >> 5);
  if (wid >= NE * HEADS) return;
  const int lane = threadIdx.x & 31;
  const int e = wid >> 2, h = wid & 3;
  const int s = src[e], d = dst[e];
  const float ea0 = eattr[e*ED+0], ea1 = eattr[e*ED+1], ea2 = eattr[e*ED+2],
              ea3 = eattr[e*ED+3], ea4 = eattr[e*ED+4];
  const float* __restrict__ xls = xl + (size_t)s * HH + h * HD;
  const float* __restrict__ xrd = xr + (size_t)d * HH + h * HD;
  const float* __restrict__ Wh  = We + (size_t)h * HD * ED;
  const float* __restrict__ ah  = att + h * HD;
  float acc = 0.0f;
  #pragma unroll
  for (int i = 0; i < 8; i++) {
    const int c = lane + 32 * i;
    const float* w = Wh + c * ED;
    float em = ea0*w[0] + ea1*w[1] + ea2*w[2] + ea3*w[3] + ea4*w[4];
    float m  = xls[c] + xrd[c] + em;
    m = m > 0.0f ? m : SLOPE * m;             // leaky_relu
    acc += m * ah[c];
  }
  for (int off = 16; off; off >>= 1) acc += __shfl_xor(acc, off, 32);
  if (lane == 0) {
    logit[wid] = acc;
    atomicMax(&mx[d * HEADS + h], f2ord(acc));
  }
}

// ---------- edge pass 2: exp + segment sum ----------
__global__ void edge_exp_k(const int* __restrict__ dst, float* __restrict__ logit,
                           const unsigned* __restrict__ mx, float* __restrict__ den)
{
  const int t = blockIdx.x * blockDim.x + threadIdx.x;
  if (t >= NE * HEADS) return;
  const int e = t >> 2, h = t & 3;
  const int d = dst[e];
  const float ex = expf(logit[t] - ord2f(mx[d * HEADS + h]));
  logit[t] = ex;
  atomicAdd(&den[d * HEADS + h], ex);
}

// ---------- edge pass 3: weighted aggregation (head-collapsed), one wave per edge ----------
__global__ __launch_bounds__(256) void edge_agg_k(
    const float* __restrict__ xl, const float* __restrict__ exl,
    const float* __restrict__ den,
    const int* __restrict__ src, const int* __restrict__ dst,
    float* __restrict__ hacc)
{
  const int e = (int)((blockIdx.x * blockDim.x + threadIdx.x) >> 5);
  if (e >= NE) return;
  const int lane = threadIdx.x & 31;
  const int s = src[e], d = dst[e];
  const float a0 = exl[e*4+0] / (den[d*4+0] + 1e-16f);
  const float a1 = exl[e*4+1] / (den[d*4+1] + 1e-16f);
  const float a2 = exl[e*4+2] / (den[d*4+2] + 1e-16f);
  const float a3 = exl[e*4+3] / (den[d*4+3] + 1e-16f);
  const float* __restrict__ xs = xl + (size_t)s * HH;
  float* __restrict__ hd = hacc + (size_t)d * HD;
  #pragma unroll
  for (int i = 0; i < 8; i++) {
    const int c = lane + 32 * i;
    const float v = a0*xs[c] + a1*xs[HD+c] + a2*xs[2*HD+c] + a3*xs[3*HD+c];
    atomicAdd(&hd[c], v);
  }
}

// ---------- node update: y = elu(hacc/4 + bias) + res  (in place in hacc) ----------
__global__ void node_y_k(float* __restrict__ hacc, const float* __restrict__ bias,
                         const float* __restrict__ res)
{
  const int t = blockIdx.x * blockDim.x + threadIdx.x;
  if (t >= BN * HD) return;
  const int c = t & (HD - 1);
  float v = hacc[t] * 0.25f + bias[c];
  v = v > 0.0f ? v : (expf(v) - 1.0f);
  hacc[t] = v + res[t];
}

// ---------- GraphNorm: one block per graph, thread = channel (deterministic) ----------
__global__ __launch_bounds__(256) void gnorm_k(
    const float* __restrict__ y, const float* __restrict__ gb,
    const float* __restrict__ gms, const float* __restrict__ gw,
    float* __restrict__ hout)
{
  const int g = blockIdx.x, c = threadIdx.x;
  const float* __restrict__ yg = y + (size_t)g * NPER * HD + c;
  float mean = 0.0f;
  for (int n = 0; n < NPER; n++) mean += yg[(size_t)n * HD];
  mean *= (1.0f / NPER);
  const float msm = gms[c] * mean;
  float var = 0.0f;
  for (int n = 0; n < NPER; n++) { float dd = yg[(size_t)n * HD] - msm; var += dd * dd; }
  var *= (1.0f / NPER);
  const float sc = gw[c] * rsqrtf(var + GEPS);
  const float bb = gb[c];
  float* __restrict__ ho = hout + (size_t)g * NPER * HD + c;
  for (int n = 0; n < NPER; n++) ho[(size_t)n * HD] = sc * (yg[(size_t)n * HD] - msm) + bb;
}

// ---------- select current node embedding per graph ----------
__global__ void select_curr_k(const float* __restrict__ x, const int* __restrict__ batch,
                              const float* __restrict__ h, float* __restrict__ curr)
{
  const int n = blockIdx.x * blockDim.x + threadIdx.x;
  if (n >= BN) return;
  if (x[n * ND] != 0.0f) {
    const int g = batch[n];
    for (int c = 0; c < HD; c++) atomicAdd(&curr[g * HD + c], h[(size_t)n * HD + c]);
  }
}

// ---------- LSTM cell: block = batch row, thread = channel ----------
__global__ __launch_bounds__(256) void lstm_k(
    const float* __restrict__ curr, const float* __restrict__ h_state,
    const float* __restrict__ c_state,
    const float* __restrict__ Wih, const float* __restrict__ Whh,
    const float* __restrict__ bih, const float* __restrict__ bhh,
    float* __restrict__ h_next, float* __restrict__ c_next)
{
  __shared__ float ce[HD], hs[HD];
  const int b = blockIdx.x, t = threadIdx.x;
  ce[t] = curr[b * HD + t];
  hs[t] = h_state[b * HD + t];
  __syncthreads();
  float g4[4];
  #pragma unroll
  for (int q = 0; q < 4; q++) {
    const int j = q * HD + t;
    float acc = bih[j] + bhh[j];
    const float* __restrict__ wi = Wih + (size_t)j * HD;
    const float* __restrict__ wh = Whh + (size_t)j * HD;
    for (int k = 0; k < HD; k++) acc += ce[k] * wi[k] + hs[k] * wh[k];
    g4[q] = acc;
  }
  const float ig = sigm(g4[0]), fg = sigm(g4[1]);
  const float gg = tanhf(g4[2]), og = sigm(g4[3]);
  const float cn = fg * c_state[b * HD + t] + ig * gg;
  c_next[b * HD + t] = cn;
  h_next[b * HD + t] = og * tanhf(cn);
}

// ---------- critic: value = relu(hn @ W1^T + b1) @ W2^T + b2 ----------
__global__ __launch_bounds__(256) void critic_k(
    const float* __restrict__ hn, const float* __restrict__ W1,
    const float* __restrict__ b1, const float* __restrict__ W2,
    const float* __restrict__ b2, float* __restrict__ value)
{
  __shared__ float hrow[HD], hid[HD];
  const int b = blockIdx.x, t = threadIdx.x;
  hrow[t] = hn[b * HD + t];
  __syncthreads();
  float acc = b1[t];
  const float* __restrict__ w = W1 + (size_t)t * HD;
  for (int k = 0; k < HD; k++) acc += hrow[k] * w[k];
  hid[t] = fmaxf(acc, 0.0f) * W2[t];
  __syncthreads();
  for (int s2 = 128; s2 > 0; s2 >>= 1) { if (t < s2) hid[t] += hid[t + s2]; __syncthreads(); }
  if (t == 0) value[b] = hid[0] + b2[0];
}

// ---------- scorer partial: t = h_next @ W1[:,HD:]^T  (16x256) ----------
__global__ __launch_bounds__(256) void tmat_k(const float* __restrict__ hn,
                                              const float* __restrict__ W1,
                                              float* __restrict__ tout)
{
  __shared__ float hrow[HD];
  const int b = blockIdx.x, t = threadIdx.x;
  hrow[t] = hn[b * HD + t];
  __syncthreads();
  const float* __restrict__ w = W1 + (size_t)t * (2 * HD) + HD;
  float acc = 0.0f;
  for (int k = 0; k < HD; k++) acc += hrow[k] * w[k];
  tout[b * HD + t] = acc;
}

// ---------- final score: one wave per node ----------
__global__ __launch_bounds__(256) void score_k(const float* __restrict__ s1,
                                               const float* __restrict__ W2,
                                               const float* __restrict__ b2,
                                               float* __restrict__ scores)
{
  const int wid = (int)((blockIdx.x * blockDim.x + threadIdx.x) >> 5);
  if (wid >= BN) return;
  const int lane = threadIdx.x & 31;
  const float* __restrict__ r = s1 + (size_t)wid * HD;
  float acc = 0.0f;
  #pragma unroll
  for (int i = 0; i < 8; i++) acc += r[lane + 32 * i] * W2[lane + 32 * i];
  for (int off = 16; off; off >>= 1) acc += __shfl_xor(acc, off, 32);
  if (lane == 0) scores[wid] = acc + b2[0];
}

// =====================================================================
extern "C" void kernel_launch(void* const* d_in, const int* in_sizes, int n_in,
                              void* d_out, int out_size, void* d_ws, size_t ws_size,
                              hipStream_t stream)
{
  (void)n_in; (void)out_size; (void)ws_size;

  // ---- input resolution (detect insertion-ordered vs fully-sorted flattening) ----
  const bool sortedTop = (in_sizes[0] == BN);   // 'batch' first => fully sorted pytree
  const float *x, *h_state, *c_state, *eattr;
  const int *edge_index, *batch;
  const float* P[44];
  if (!sortedTop) {
    x = (const float*)d_in[0]; edge_index = (const int*)d_in[1];
    h_state = (const float*)d_in[2]; c_state = (const float*)d_in[3];
    batch = (const int*)d_in[4]; eattr = (const float*)d_in[5];
    for (int i = 0; i < 44; i++) P[i] = (const float*)d_in[6 + i];
  } else {
    batch = (const int*)d_in[0]; c_state = (const float*)d_in[1];
    eattr = (const float*)d_in[2]; edge_index = (const int*)d_in[3];
    h_state = (const float*)d_in[4];
    for (int i = 0; i < 44; i++) P[i] = (const float*)d_in[5 + i];
    x = (const float*)d_in[49];
  }
  const int* src = edge_index;
  const int* dst = edge_index + NE;
  // params (sorted keys): conv{l}: We,Wl,Wr,att,bias,bl,br
  #define CONV(l, f) P[(l) * 7 + (f)]   // f: 0=We 1=Wl 2=Wr 3=att 4=bias 5=bl 6=br
  const float *crW1 = P[21], *crW2 = P[22], *crb1 = P[23], *crb2 = P[24];
  #define GN(l, f) P[25 + (l) * 3 + (f)]  // f: 0=b 1=ms 2=w
  const float *Whh = P[34], *Wih = P[35], *bhh = P[36], *bih = P[37];
  const float *projW = P[38], *projb = P[39];
  const float *scW1 = P[40], *scW2 = P[41], *scb1 = P[42], *scb2 = P[43];

  // ---- workspace carve-up (floats) ----
  float* ws = (float*)d_ws;
  float*    xl    = ws;                               // BN*HH
  float*    xr    = xl   + (size_t)BN * HH;           // BN*HH
  float*    exl   = xr   + (size_t)BN * HH;           // NE*HEADS (logits then exp)
  unsigned* mx    = (unsigned*)(exl + (size_t)NE * HEADS); // BN*HEADS
  float*    den   = (float*)mx + (size_t)BN * HEADS;  // BN*HEADS
  float*    hacc  = den  + (size_t)BN * HEADS;        // BN*HD (agg / y / s1)
  float*    hA    = hacc + (size_t)BN * HD;           // BN*HD
  float*    hB    = hA   + (size_t)BN * HD;           // BN*HD
  float*    h_res = hB   + (size_t)BN * HD;           // BN*HD
  float*    curr  = h_res + (size_t)BN * HD;          // BG*HD
  float*    tbuf  = curr + (size_t)BG * HD;           // BG*HD

  float* out_scores = (float*)d_out;
  float* out_hn     = out_scores + BN;
  float* out_cn     = out_hn + BG * HD;
  float* out_val    = out_cn + BG * HD;

  const int GEMM_N1024_BLOCKS = (BN / 16) * (HH / 64) / 8;  // 2000
  const int GEMM_N256_BLOCKS  = (BN / 16) * (HD / 64) / 8;  // 500

  // ---- GNN layers ----
  for (int l = 0; l < 3; l++) {
    const float* Ain = (l == 0) ? x : ((l == 1) ? hA : hB);
    const int Kl = (l == 0) ? ND : HD;
    gemm_wmma_f32<<<GEMM_N1024_BLOCKS, 256, 0, stream>>>(
        Ain, Kl, CONV(l, 1), Kl, CONV(l, 5), nullptr, nullptr, xl, BN, HH, Kl, 0);
    gemm_wmma_f32<<<GEMM_N1024_BLOCKS, 256, 0, stream>>>(
        Ain, Kl, CONV(l, 2), Kl, CONV(l, 6), nullptr, nullptr, xr, BN, HH, Kl, 0);
    if (l == 0)
      gemm_wmma_f32<<<GEMM_N256_BLOCKS, 256, 0, stream>>>(
          x, ND, projW, ND, projb, nullptr, nullptr, h_res, BN, HD, ND, 0);

    hipMemsetAsync(mx,   0, (size_t)BN * HEADS * 4, stream);
    hipMemsetAsync(den,  0, (size_t)BN * HEADS * 4, stream);
    hipMemsetAsync(hacc, 0, (size_t)BN * HD * 4,   stream);

    edge_logits_k<<<(NE * HEADS * 32) / 256, 256, 0, stream>>>(
        xl, xr, eattr, CONV(l, 0), CONV(l, 3), src, dst, exl, mx);
    edge_exp_k<<<(NE * HEADS) / 256, 256, 0, stream>>>(dst, exl, mx, den);
    edge_agg_k<<<(NE * 32) / 256, 256, 0, stream>>>(xl, exl, den, src, dst, hacc);

    node_y_k<<<(BN * HD) / 256, 256, 0, stream>>>(
        hacc, CONV(l, 4), (l == 0) ? h_res : Ain);
    float* hout = (l == 1) ? hB : hA;
    gnorm_k<<<BG, 256, 0, stream>>>(hacc, GN(l, 0), GN(l, 1), GN(l, 2), hout);
  }
  // final node embeddings in hA

  // ---- LSTM head ----
  hipMemsetAsync(curr, 0, (size_t)BG * HD * 4, stream);
  select_curr_k<<<(BN + 255) / 256, 256, 0, stream>>>(x, batch, hA, curr);
  lstm_k<<<BG, 256, 0, stream>>>(curr, h_state, c_state, Wih, Whh, bih, bhh,
                                 out_hn, out_cn);
  critic_k<<<BG, 256, 0, stream>>>(out_hn, crW1, crb1, crW2, crb2, out_val);

  // ---- scorer ----
  tmat_k<<<BG, 256, 0, stream>>>(out_hn, scW1, tbuf);
  gemm_wmma_f32<<<GEMM_N256_BLOCKS, 256, 0, stream>>>(
      hA, HD, scW1, 2 * HD, scb1, tbuf, batch, hacc /* s1 */, BN, HD, HD, 1);
  score_k<<<(BN * 32) / 256, 256, 0, stream>>>(hacc, scW2, scb2, out_scores);
}